// Rule_47450798686483
// MI455X (gfx1250) — compile-verified
//
#include <hip/hip_runtime.h>
#include <cstdint>
#include <cstddef>

// Problem constants (match reference: B,H,W = 4,2048,2048, RADIUS=1)
#define B_  4
#define H_  2048
#define W_  2048
#define HW_ (H_ * W_)
#define TW  1024          // columns per block tile
#define NT_ (W_ / TW)     // tiles per row == 2

typedef float v4f __attribute__((ext_vector_type(4)));
typedef int   v4i_vs __attribute__((vector_size(16)));
typedef __attribute__((address_space(1))) v4i_vs* as1_v4i_ptr;
typedef __attribute__((address_space(3))) v4i_vs* as3_v4i_ptr;

__device__ __forceinline__ v4f ntload4(const float* p) {
    return __builtin_nontemporal_load((const v4f*)p);
}
__device__ __forceinline__ void ntstore4(float* p, v4f v) {
    __builtin_nontemporal_store(v, (v4f*)p);
}

// 16B global -> LDS copy. Prefer the CDNA5 async-DMA path (ASYNCcnt) when the
// toolchain exposes the builtin; otherwise fall back to a VGPR-mediated copy.
__device__ __forceinline__ void copy16_g2lds(const float* g, float* l) {
#if defined(__AMDGCN__) && __has_builtin(__builtin_amdgcn_global_load_async_to_lds_b128)
    v4i_vs* gg = (v4i_vs*)g;   // pointee reinterpret (+const removal), generic AS
    v4i_vs* ll = (v4i_vs*)l;
    __builtin_amdgcn_global_load_async_to_lds_b128(
        (as1_v4i_ptr)gg,       // pure addrspacecast generic -> global
        (as3_v4i_ptr)ll,       // pure addrspacecast generic -> LDS
        0 /*offset*/, 0 /*cpol*/);
#else
    *(v4f*)l = *(const v4f*)g;
#endif
}

__device__ __forceinline__ void wait_g2lds() {
#if defined(__AMDGCN__) && __has_builtin(__builtin_amdgcn_s_wait_asynccnt)
    __builtin_amdgcn_s_wait_asynccnt(0);
#elif defined(__AMDGCN__)
    asm volatile("s_wait_asynccnt 0" ::: "memory");
#endif
}

__global__ __launch_bounds__(256) void ising_sweep(const float* __restrict__ x,
                                                   const float* __restrict__ rnd,
                                                   const float* __restrict__ dro,
                                                   float* __restrict__ out) {
    __shared__ __align__(16) float l_up[TW];
    __shared__ __align__(16) float l_c [TW];
    __shared__ __align__(16) float l_dn[TW];
    __shared__ float l_halo[2];

    const int tid = threadIdx.x;
    unsigned t  = blockIdx.x;                 // (bi*H + i)*NT_ + tx
    const int tx = t & (NT_ - 1);  t >>= 1;   // NT_ == 2
    const int i  = t & (H_ - 1);   t >>= 11;  // H_ == 2048
    const int bi = (int)t;

    const size_t j0   = (size_t)tx * TW;
    const float* sP   = x + (size_t)bi * 3 * HW_;
    const float* tP   = sP + HW_;
    const float* bP   = sP + 2 * (size_t)HW_;
    const size_t rowO = (size_t)i * W_;
    const size_t upO  = (size_t)((i == 0)      ? (H_ - 1) : (i - 1)) * W_;
    const size_t dnO  = (size_t)((i == H_ - 1) ? 0        : (i + 1)) * W_;
    const int    c    = tid * 4;

    // Stage the three spin rows for this tile into LDS via async DMA.
    copy16_g2lds(sP + upO  + j0 + c, &l_up[c]);
    copy16_g2lds(sP + rowO + j0 + c, &l_c [c]);
    copy16_g2lds(sP + dnO  + j0 + c, &l_dn[c]);

    // Periodic column halo for the center row (2 scalars per tile).
    if (tid == 0) {
        const size_t jl = (j0 == 0)        ? (size_t)(W_ - 1) : (j0 - 1);
        const size_t jr = (j0 + TW == W_)  ? (size_t)0        : (j0 + TW);
        l_halo[0] = sP[rowO + jl];
        l_halo[1] = sP[rowO + jr];
    }

    // Single-use streams: non-temporal loads (don't pollute L2, which is
    // holding the reused s-plane + dropout plane). Overlaps the async DMA.
    const v4f tr = ntload4(tP + rowO + j0 + c);
    const v4f bb = ntload4(bP + rowO + j0 + c);
    const v4f rd = ntload4(rnd + (size_t)bi * HW_ + rowO + j0 + c);
    const v4f dq = *(const v4f*)(dro + rowO + j0 + c);   // reused 4x (per batch): keep RT

    wait_g2lds();
    __syncthreads();

    const v4f sc = *(const v4f*)&l_c [c];
    const v4f su = *(const v4f*)&l_up[c];
    const v4f sd = *(const v4f*)&l_dn[c];
    const float sl = (tid == 0)   ? l_halo[0] : l_c[c - 1];
    const float sr = (tid == 255) ? l_halo[1] : l_c[c + 4];

    v4f os, ot;
#pragma unroll
    for (int k = 0; k < 4; ++k) {
        const float s  = sc[k];
        const float L  = (k == 0) ? sl : sc[k - 1];
        const float R  = (k == 3) ? sr : sc[k + 1];
        const float Js = su[k] + sd[k] + L + R;
        const float de = 2.0f * s * Js;
        const float p  = (de <= 0.0f) ? 1.0f : expf(-de * bb[k]);
        const bool  fl = (rd[k] < p) && (dq[k] > 0.5f);
        os[k] = fl ? -s : s;
        ot[k] = 0.9f * tr[k] + 0.1f * s;
    }

    float* oS = out + (size_t)bi * 3 * HW_;
    ntstore4(oS +                  rowO + j0 + c, os);   // s * flip
    ntstore4(oS + HW_ +            rowO + j0 + c, ot);   // tr_new
    ntstore4(oS + 2 * (size_t)HW_ + rowO + j0 + c, bb);  // b passthrough
}

extern "C" void kernel_launch(void* const* d_in, const int* in_sizes, int n_in,
                              void* d_out, int out_size, void* d_ws, size_t ws_size,
                              hipStream_t stream) {
    (void)in_sizes; (void)n_in; (void)d_ws; (void)ws_size; (void)out_size;
    const float* x   = (const float*)d_in[0];   // (B,3,H,W)
    const float* rnd = (const float*)d_in[1];   // (B,1,H,W)
    const float* dro = (const float*)d_in[2];   // (H,W)
    float*       out = (float*)d_out;           // (B,3,H,W)

    const int grid = B_ * H_ * NT_;             // 16384 blocks of 256 threads
    ising_sweep<<<grid, 256, 0, stream>>>(x, rnd, dro, out);
}